// HopfONN_8126078124467
// MI455X (gfx1250) — compile-verified
//
#include <hip/hip_runtime.h>

#define NN 16
#define BATCH 524288
#define STEPS 20
#define ALPHA_C 1.0f
#define DT_C 0.5f

typedef float v2f __attribute__((ext_vector_type(2)));
typedef float v4f __attribute__((ext_vector_type(4)));
typedef float v8f __attribute__((ext_vector_type(8)));

// Half-wave swap via LDS crossbar (used only in the epilogue head reduction).
__device__ __forceinline__ float swap16(float x) {
    return __int_as_float(__builtin_amdgcn_ds_swizzle(__float_as_int(x), 0x401F));
}

// v_permlane16_swap_b32: single-op exchange of lane halves between two VGPRs.
// Returns x = {a.lanes0-15, b.lanes0-15}, y = {a.lanes16-31, b.lanes16-31}.
__device__ __forceinline__ void plswap(float a, float b, float& x, float& y) {
    auto pr = __builtin_amdgcn_permlane16_swap(__float_as_uint(a),
                                               __float_as_uint(b),
                                               false, false);
    x = __uint_as_float(((const unsigned*)&pr)[0]);
    y = __uint_as_float(((const unsigned*)&pr)[1]);
}

// One dynamics evaluation: dz = ((alpha - |z|^2) + i) z + W z  for 16 batch
// elements per wave, state held in WMMA B-layout (lane = batch, vgpr = comp).
// A0 = rows 0..15 of the real 32x32 lift (real outputs), A1 = rows 16..31.
__device__ __forceinline__ void dyn(const v2f* __restrict__ A0,
                                    const v2f* __restrict__ A1,
                                    const float* __restrict__ zin,
                                    float* __restrict__ dz) {
    v8f accR = {0.f, 0.f, 0.f, 0.f, 0.f, 0.f, 0.f, 0.f};
    v8f accI = {0.f, 0.f, 0.f, 0.f, 0.f, 0.f, 0.f, 0.f};
#pragma unroll
    for (int q = 0; q < 8; ++q) {
        v2f bop;
        bop[0] = zin[2 * q];
        bop[1] = zin[2 * q + 1];
        accR = __builtin_amdgcn_wmma_f32_16x16x4_f32(
            false, A0[q], false, bop, (short)0, accR, false, false);
        accI = __builtin_amdgcn_wmma_f32_16x16x4_f32(
            false, A1[q], false, bop, (short)0, accI, false, false);
    }
    // D-layout -> B-layout: 8 x v_permlane16_swap (was 16 ds_swizzle + 16 sel).
    float wz[16];
    plswap(accR[0], accR[2], wz[0], wz[4]);
    plswap(accR[1], accR[3], wz[1], wz[5]);
    plswap(accR[4], accR[6], wz[2], wz[6]);
    plswap(accR[5], accR[7], wz[3], wz[7]);
    plswap(accI[0], accI[2], wz[8],  wz[12]);
    plswap(accI[1], accI[3], wz[9],  wz[13]);
    plswap(accI[4], accI[6], wz[10], wz[14]);
    plswap(accI[5], accI[7], wz[11], wz[15]);
#pragma unroll
    for (int v = 0; v < 8; ++v) {
        float re = zin[v], im = zin[v + 8];
        float t = ALPHA_C - (re * re + im * im);
        dz[v]     = __builtin_fmaf(t, re, wz[v]) - im;      // t*re - im + (Wz)re
        dz[v + 8] = __builtin_fmaf(t, im, wz[v + 8]) + re;  // t*im + re + (Wz)im
    }
}

__global__ void __launch_bounds__(256, 1)
hopf_onn_kernel(const float* __restrict__ x, const float* __restrict__ Wr,
                const float* __restrict__ Wi, const float* __restrict__ hw,
                const float* __restrict__ hb, float* __restrict__ out) {
    const int lane = threadIdx.x & 31;
    const int wave = threadIdx.x >> 5;
    const int hh = lane >> 4;        // lane half
    const bool lo = (hh == 0);
    const int bl = lane & 15;        // batch row within tile
    const int tile = blockIdx.x * 8 + wave;
    const int b = tile * 16 + bl;

    // ---- Load 32x32 lifted weight matrix into WMMA A-layout registers ----
    // A tile t, slice q: vgpr e holds M[row = 16t + bl][k = 4q + 2h + e].
    v2f A0[8], A1[8];
#pragma unroll
    for (int q = 0; q < 8; ++q) {
#pragma unroll
        for (int e = 0; e < 2; ++e) {
            int k = 4 * q + 2 * hh + e;
            float m0, m1;
            if (q < 4) {                // k < 16: columns hit [Wr; Wi]
                m0 = Wr[bl * NN + k];
                m1 = Wi[bl * NN + k];
            } else {                    // k >= 16: columns hit [-Wi; Wr]
                int kk = k - 16;
                m0 = -Wi[bl * NN + kk];
                m1 =  Wr[bl * NN + kk];
            }
            A0[q][e] = m0;
            A1[q][e] = m1;
        }
    }

    // ---- Phase-encode initial state z0 = exp(i*2pi*log2(max(x,1))/11) ----
    // v_log_f32 is log2; v_sin/cos_f32 take input in revolutions (x/2pi).
    float z[16];
#pragma unroll
    for (int v = 0; v < 8; ++v) {
        int j = 4 * (v >> 1) + 2 * hh + (v & 1);
        float xv = x[(size_t)b * NN + j];
        xv = fmaxf(xv, 1.0f);
        float xn = __builtin_amdgcn_logf(xv) * (1.0f / 11.0f);
        z[v]     = __builtin_amdgcn_cosf(xn);
        z[v + 8] = __builtin_amdgcn_sinf(xn);
    }

    // ---- 19 RK4 steps, h = DT*STEPS/(STEPS-1) ----
    const float hs = DT_C * (float)STEPS / (float)(STEPS - 1);
    float acc[16], zt[16], k[16];
#pragma unroll 1
    for (int s = 0; s < STEPS - 1; ++s) {
        dyn(A0, A1, z, k);                           // k1
#pragma unroll
        for (int v = 0; v < 16; ++v) {
            acc[v] = k[v];
            zt[v] = __builtin_fmaf(0.5f * hs, k[v], z[v]);
        }
        dyn(A0, A1, zt, k);                          // k2
#pragma unroll
        for (int v = 0; v < 16; ++v) {
            acc[v] = __builtin_fmaf(2.0f, k[v], acc[v]);
            zt[v] = __builtin_fmaf(0.5f * hs, k[v], z[v]);
        }
        dyn(A0, A1, zt, k);                          // k3
#pragma unroll
        for (int v = 0; v < 16; ++v) {
            acc[v] = __builtin_fmaf(2.0f, k[v], acc[v]);
            zt[v] = __builtin_fmaf(hs, k[v], z[v]);
        }
        dyn(A0, A1, zt, k);                          // k4
#pragma unroll
        for (int v = 0; v < 16; ++v) {
            acc[v] += k[v];
            z[v] = __builtin_fmaf(hs * (1.0f / 6.0f), acc[v], z[v]);
        }
    }

    // ---- Head: out[b,o] = sum_s feat[s]*hw[o,s] + hb[o], feat=[Re z; Im z] ----
    float o[4];
#pragma unroll
    for (int oo = 0; oo < 4; ++oo) {
        float p = 0.f;
#pragma unroll
        for (int v = 0; v < 8; ++v) {
            int j = 4 * (v >> 1) + 2 * hh + (v & 1);
            p = __builtin_fmaf(z[v],     hw[oo * 32 + j],      p);
            p = __builtin_fmaf(z[v + 8], hw[oo * 32 + 16 + j], p);
        }
        p += swap16(p);   // partner half-lane holds the other 8 components
        o[oo] = p + hb[oo];
    }
    if (lo) {
        v4f r;
        r[0] = o[0]; r[1] = o[1]; r[2] = o[2]; r[3] = o[3];
        *(v4f*)(out + (size_t)b * 4) = r;
    }
}

extern "C" void kernel_launch(void* const* d_in, const int* in_sizes, int n_in,
                              void* d_out, int out_size, void* d_ws, size_t ws_size,
                              hipStream_t stream) {
    const float* x  = (const float*)d_in[0];
    const float* Wr = (const float*)d_in[1];
    const float* Wi = (const float*)d_in[2];
    const float* hw = (const float*)d_in[3];
    const float* hb = (const float*)d_in[4];
    float* out = (float*)d_out;
    // 256 threads = 8 waves x 16 batch rows -> 128 elements per block.
    dim3 grid(BATCH / 128);
    hopf_onn_kernel<<<grid, 256, 0, stream>>>(x, Wr, Wi, hw, hb, out);
}